// DeformableAttention3D_5583457485252
// MI455X (gfx1250) — compile-verified
//
#include <hip/hip_runtime.h>
#include <hip/hip_bf16.h>

typedef __attribute__((ext_vector_type(2))) float v2f;
typedef __attribute__((ext_vector_type(8))) float v8f;

#define EMBED 256
#define HEADS 8
#define HD    32
#define LVLS  4
#define NR1   4
#define NR2   8
#define DBINS 64
#define NCAM  6
#define NQ    2000

// ---------------------------------------------------------------------------
// Software-pipelined K=256 WMMA tile: A from LDS row (stride-257 layout),
// B from a contiguous 256-float column of the transposed weight matrix.
// 8-deep rotating operand buffers -> loads stay ~8 WMMAs ahead of use.
// ---------------------------------------------------------------------------
__device__ __forceinline__ void wmma_k256_tile(const float* __restrict__ ap,
                                               const float* __restrict__ bp,
                                               int half, v8f& acc0, v8f& acc1)
{
    const int off = 2 * half;      // lanes 0-15: K{+0,+1}; lanes 16-31: K{+2,+3}
    v2f areg[8], breg[8];
#pragma unroll
    for (int j = 0; j < 8; ++j) {
        breg[j] = *(const v2f*)(bp + j * 4 + off);   // global_load_b64
        areg[j].x = ap[j * 4 + off];                 // ds_load_2addr_b32
        areg[j].y = ap[j * 4 + off + 1];
    }
#pragma unroll
    for (int s = 0; s < 64; s += 8) {                // 64 K-steps of 4
        if ((s & 15) == 0) __builtin_prefetch(bp + s * 4 + 128, 0, 1);
#pragma unroll
        for (int j = 0; j < 8; ++j) {
            const v2f a = areg[j];
            const v2f b = breg[j];
            if (s + 8 < 64) {                        // reload 8 steps ahead
                const int kn = (s + j + 8) * 4 + off;
                breg[j] = *(const v2f*)(bp + kn);
                areg[j].x = ap[kn];
                areg[j].y = ap[kn + 1];
            }
            if (j & 1)
                acc1 = __builtin_amdgcn_wmma_f32_16x16x4_f32(false, a, false, b,
                                                             (short)0, acc1, false, false);
            else
                acc0 = __builtin_amdgcn_wmma_f32_16x16x4_f32(false, a, false, b,
                                                             (short)0, acc0, false, false);
        }
    }
}

// ---------------------------------------------------------------------------
// Kernel 0: tiled transpose (with optional zero-pad of the N dimension).
// src: [256][N] row-major  ->  dst: [Npad][256] row-major (dst rows >= N are 0)
// ---------------------------------------------------------------------------
__global__ void __launch_bounds__(256)
transpose_pad_kernel(const float* __restrict__ W, float* __restrict__ WT,
                     int N, int Npad)
{
    __shared__ float tile[32][33];
    const int tx = threadIdx.x & 31;
    const int ty = threadIdx.x >> 5;          // 0..7
    const int n0 = blockIdx.x * 32;
    const int k0 = blockIdx.y * 32;
#pragma unroll
    for (int j = 0; j < 4; ++j) {
        const int k = k0 + ty + j * 8;
        const int n = n0 + tx;
        tile[ty + j * 8][tx] = (n < N) ? W[(size_t)k * N + n] : 0.f;
    }
    __syncthreads();
#pragma unroll
    for (int j = 0; j < 4; ++j) {
        const int n = n0 + ty + j * 8;
        if (n < Npad) WT[(size_t)n * 256 + (k0 + tx)] = tile[tx][ty + j * 8];
    }
}

// ---------------------------------------------------------------------------
// Kernel 1: fused query projections (delta / attn logits / off3-padded).
// fp32 WMMA 16x16x4; A-tile in LDS; B from transposed weights.
// ---------------------------------------------------------------------------
__global__ void __launch_bounds__(128)
gemm_qproj_kernel(const float* __restrict__ q,
                  const float* __restrict__ WTd, const float* __restrict__ bd,
                  const float* __restrict__ WTa, const float* __restrict__ ba,
                  const float* __restrict__ WT3, const float* __restrict__ bo,
                  float* __restrict__ delta, float* __restrict__ attn,
                  float* __restrict__ off3)
{
    __shared__ float Alds[16][257];
    const int tid  = threadIdx.x;
    const int row0 = blockIdx.x * 16;
    for (int i = tid; i < 16 * 256; i += 128) {
        int r = i >> 8, c = i & 255;
        Alds[r][c] = q[(size_t)(row0 + r) * EMBED + c];
    }
    __syncthreads();

    const int gw = blockIdx.y * 4 + (tid >> 5);   // column-tile id, 0..256
    if (gw > 256) return;

    const float* WT; const float* bias; float* outp; int ldo; int col0; bool is3 = false;
    if (gw < 192)      { WT = WTd; bias = bd; outp = delta; ldo = 3072; col0 = gw * 16; }
    else if (gw < 256) { WT = WTa; bias = ba; outp = attn;  ldo = 1024; col0 = (gw - 192) * 16; }
    else               { WT = WT3; bias = bo; outp = off3;  ldo = 16;   col0 = 0; is3 = true; }

    const int lane = tid & 31;
    const int half = lane >> 4;
    const int l16  = lane & 15;
    const int col  = col0 + l16;

    const float* ap = &Alds[l16][0];
    const float* bp = WT + (size_t)col * 256;     // contiguous K stream per lane

    v8f acc0 = {0.f, 0.f, 0.f, 0.f, 0.f, 0.f, 0.f, 0.f};
    v8f acc1 = acc0;
    wmma_k256_tile(ap, bp, half, acc0, acc1);

    const float bv = is3 ? (l16 < 12 ? bias[l16] : 0.f) : bias[col];
#pragma unroll
    for (int r = 0; r < 8; ++r) {
        const int row = row0 + r + 8 * half;      // C/D layout: VGPR r -> M=r (+8 hi half)
        outp[(size_t)row * ldo + col] = acc0[r] + acc1[r] + bv;
    }
}

// ---------------------------------------------------------------------------
// Kernel 2: softmax over 128 logits per (query, head). One wave per row.
// ---------------------------------------------------------------------------
__global__ void __launch_bounds__(128)
softmax_kernel(float* __restrict__ attn)
{
    const int w = blockIdx.x * 4 + (threadIdx.x >> 5);
    if (w >= NQ * HEADS) return;
    const int lane = threadIdx.x & 31;
    float* p = attn + (size_t)w * 128;
    float v0 = p[lane], v1 = p[lane + 32], v2 = p[lane + 64], v3 = p[lane + 96];
    float m = fmaxf(fmaxf(v0, v1), fmaxf(v2, v3));
    for (int off = 16; off > 0; off >>= 1) m = fmaxf(m, __shfl_xor(m, off, 32));
    v0 = expf(v0 - m); v1 = expf(v1 - m); v2 = expf(v2 - m); v3 = expf(v3 - m);
    float s = v0 + v1 + v2 + v3;
    for (int off = 16; off > 0; off >>= 1) s += __shfl_xor(s, off, 32);
    const float inv = 1.f / s;
    p[lane] = v0 * inv; p[lane + 32] = v1 * inv;
    p[lane + 64] = v2 * inv; p[lane + 96] = v3 * inv;
}

// ---------------------------------------------------------------------------
// Kernel 3: project anchors through lidar2img -> (u, v, dn, valid)
// off3 buffer is padded: [NQ][16], cols 0..11 valid.
// ---------------------------------------------------------------------------
__global__ void proj_kernel(const float* __restrict__ gm, const float* __restrict__ off3,
                            const float* __restrict__ l2i,
                            const int* __restrict__ Hop, const int* __restrict__ Wop,
                            float* __restrict__ refuvd, float* __restrict__ valid)
{
    const int i = blockIdx.x * blockDim.x + threadIdx.x;
    if (i >= NCAM * NQ * NR1) return;
    const int r1 = i % NR1;
    const int n  = (i / NR1) % NQ;
    const int c  = i / (NR1 * NQ);
    const float x = gm[n * 3 + 0] + off3[(size_t)n * 16 + r1 * 3 + 0];
    const float y = gm[n * 3 + 1] + off3[(size_t)n * 16 + r1 * 3 + 1];
    const float z = gm[n * 3 + 2] + off3[(size_t)n * 16 + r1 * 3 + 2];
    const float* M = l2i + c * 16;
    const float px = M[0] * x + M[1] * y + M[2]  * z + M[3];
    const float py = M[4] * x + M[5] * y + M[6]  * z + M[7];
    const float pz = M[8] * x + M[9] * y + M[10] * z + M[11];
    const float depth = fmaxf(pz, 1e-5f);
    const float Wo = (float)(*Wop), Ho = (float)(*Hop);
    const float u  = px / depth / Wo;
    const float v  = py / depth / Ho;
    const float dn = fminf(fmaxf((depth - 1.0f) * (1.0f / 59.0f), 0.f), 1.f);
    const bool ok = (u >= 0.f) && (u <= 1.f) && (v >= 0.f) && (v <= 1.f) && (depth > 1.0f);
    const size_t o = ((size_t)c * NQ + n) * NR1 + r1;
    refuvd[o * 3 + 0] = u; refuvd[o * 3 + 1] = v; refuvd[o * 3 + 2] = dn;
    valid[o] = ok ? 1.f : 0.f;
}

// ---------------------------------------------------------------------------
// Kernel 4: deformable sampling. Block = one query (256 thr = 8 waves).
// Wave = head, lane = channel (HD==32==wave32). attn/delta/ref staged in LDS.
// All gathers are L2-resident (feats+dmaps ~30MB << 192MB L2).
// ---------------------------------------------------------------------------
__global__ void __launch_bounds__(256)
sample_kernel(const float* __restrict__ f0, const float* __restrict__ f1,
              const float* __restrict__ f2, const float* __restrict__ f3,
              const float* __restrict__ d0, const float* __restrict__ d1,
              const float* __restrict__ d2, const float* __restrict__ d3,
              const float* __restrict__ refuvd, const float* __restrict__ valid,
              const float* __restrict__ attn, const float* __restrict__ delta,
              float* __restrict__ accbuf)
{
    __shared__ float s_delta[HEADS * LVLS * NR1 * NR2 * 3];  // 3072
    __shared__ float s_attn[HEADS * LVLS * NR1 * NR2];       // 1024
    __shared__ float s_ref[NCAM * NR1 * 3];
    __shared__ float s_val[NCAM * NR1];
    const int n = blockIdx.x;
    const int tid = threadIdx.x;
    for (int i = tid; i < 3072; i += 256) s_delta[i] = delta[(size_t)n * 3072 + i];
    for (int i = tid; i < 1024; i += 256) s_attn[i]  = attn[(size_t)n * 1024 + i];
    for (int i = tid; i < NCAM * NR1; i += 256) {
        const int c = i / NR1, r = i % NR1;
        const size_t o = ((size_t)c * NQ + n) * NR1 + r;
        s_val[i] = valid[o];
        s_ref[i * 3 + 0] = refuvd[o * 3 + 0];
        s_ref[i * 3 + 1] = refuvd[o * 3 + 1];
        s_ref[i * 3 + 2] = refuvd[o * 3 + 2];
    }
    __syncthreads();

    const int h  = tid >> 5;
    const int ch = tid & 31;
    const float* fl[4] = {f0, f1, f2, f3};
    const float* dl[4] = {d0, d1, d2, d3};
    const int Hs[4] = {32, 16, 8, 4};
    const int Ws[4] = {88, 44, 22, 11};

    float acc = 0.f;
    for (int c = 0; c < NCAM; ++c) {
        for (int r1 = 0; r1 < NR1; ++r1) {
            if (s_val[c * NR1 + r1] == 0.f) continue;     // valid-mask zeroes aw
            const float bu = s_ref[(c * NR1 + r1) * 3 + 0];
            const float bv = s_ref[(c * NR1 + r1) * 3 + 1];
            const float bd = s_ref[(c * NR1 + r1) * 3 + 2];
            for (int l = 0; l < LVLS; ++l) {
                const int Hl = Hs[l], Wl = Ws[l], HW = Hl * Wl;
                const float* Fb = fl[l] + (size_t)(c * EMBED + h * HD + ch) * HW;
                const float* Db = dl[l] + (size_t)c * DBINS * HW;
                for (int r2 = 0; r2 < NR2; ++r2) {
                    const int di = (((h * LVLS + l) * NR1 + r1) * NR2 + r2) * 3;
                    const float u  = fminf(fmaxf(bu + s_delta[di + 0], 0.f), 1.f);
                    const float v  = fminf(fmaxf(bv + s_delta[di + 1], 0.f), 1.f);
                    const float dd = fminf(fmaxf(bd + s_delta[di + 2], 0.f), 1.f);
                    const float aw = s_attn[h * 128 + (l * NR1 + r1) * NR2 + r2];
                    const float x = u * (float)Wl - 0.5f;
                    const float y = v * (float)Hl - 0.5f;
                    const float x0f = floorf(x), y0f = floorf(y);
                    const float wx = x - x0f, wy = y - y0f;
                    const int x0 = (int)x0f, y0 = (int)y0f;
                    const float dc = dd * (float)(DBINS - 1);
                    int db = (int)floorf(dc);
                    db = db < 0 ? 0 : (db > DBINS - 2 ? DBINS - 2 : db);
                    const float wh = dc - (float)db;
                    float f = 0.f, sc = 0.f;
#pragma unroll
                    for (int corner = 0; corner < 4; ++corner) {
                        const int dy = corner >> 1, dx = corner & 1;
                        const int xi = x0 + dx, yi = y0 + dy;
                        if (xi < 0 || xi >= Wl || yi < 0 || yi >= Hl) continue;
                        const float w = (dy ? wy : 1.f - wy) * (dx ? wx : 1.f - wx);
                        const int idx = yi * Wl + xi;
                        f  += w * Fb[idx];
                        sc += w * ((1.f - wh) * Db[db * HW + idx] +
                                   wh * Db[(db + 1) * HW + idx]);
                    }
                    acc += f * sc * aw;
                }
            }
        }
    }
    accbuf[(size_t)n * EMBED + h * HD + ch] = acc;
}

// ---------------------------------------------------------------------------
// Kernel 5: out = (acc / NC) @ W_out + b_out  (2000x256x256, fp32 WMMA)
// ---------------------------------------------------------------------------
__global__ void __launch_bounds__(128)
gemm_out_kernel(const float* __restrict__ accbuf, const float* __restrict__ WTo,
                const float* __restrict__ bout, float* __restrict__ out)
{
    __shared__ float Alds[16][257];
    const int tid  = threadIdx.x;
    const int row0 = blockIdx.x * 16;
    const float scale = 1.f / (float)NCAM;
    for (int i = tid; i < 16 * 256; i += 128) {
        int r = i >> 8, c = i & 255;
        Alds[r][c] = accbuf[(size_t)(row0 + r) * EMBED + c] * scale;
    }
    __syncthreads();

    const int gw   = blockIdx.y * 4 + (tid >> 5);
    const int lane = tid & 31;
    const int half = lane >> 4;
    const int l16  = lane & 15;
    const int col  = gw * 16 + l16;

    const float* ap = &Alds[l16][0];
    const float* bp = WTo + (size_t)col * 256;

    v8f acc0 = {0.f, 0.f, 0.f, 0.f, 0.f, 0.f, 0.f, 0.f};
    v8f acc1 = acc0;
    wmma_k256_tile(ap, bp, half, acc0, acc1);

    const float bv = bout[col];
#pragma unroll
    for (int r = 0; r < 8; ++r)
        out[(size_t)(row0 + r + 8 * half) * EMBED + col] = acc0[r] + acc1[r] + bv;
}

// ---------------------------------------------------------------------------
extern "C" void kernel_launch(void* const* d_in, const int* in_sizes, int n_in,
                              void* d_out, int out_size, void* d_ws, size_t ws_size,
                              hipStream_t stream)
{
    const float* query = (const float*)d_in[0];
    const float* gm    = (const float*)d_in[1];
    const float* f0    = (const float*)d_in[2];
    const float* f1    = (const float*)d_in[3];
    const float* f2    = (const float*)d_in[4];
    const float* f3    = (const float*)d_in[5];
    const float* dm0   = (const float*)d_in[6];
    const float* dm1   = (const float*)d_in[7];
    const float* dm2   = (const float*)d_in[8];
    const float* dm3   = (const float*)d_in[9];
    const float* l2i   = (const float*)d_in[10];
    const float* Woff3 = (const float*)d_in[11];
    const float* boff3 = (const float*)d_in[12];
    const float* Wuvd  = (const float*)d_in[13];
    const float* buvd  = (const float*)d_in[14];
    const float* Wattn = (const float*)d_in[15];
    const float* battn = (const float*)d_in[16];
    const float* Wout  = (const float*)d_in[17];
    const float* bout  = (const float*)d_in[18];
    const int*   Hop   = (const int*)d_in[19];
    const int*   Wop   = (const int*)d_in[20];

    float* ws    = (float*)d_ws;
    float* off3  = ws;                                   // NQ * 16 (padded)
    float* delta = off3  + (size_t)NQ * 16;              // NQ * 3072
    float* attn  = delta + (size_t)NQ * 3072;            // NQ * 1024
    float* ref   = attn  + (size_t)NQ * 1024;            // NCAM*NQ*NR1*3
    float* val   = ref   + (size_t)NCAM * NQ * NR1 * 3;  // NCAM*NQ*NR1
    float* accb  = val   + (size_t)NCAM * NQ * NR1;      // NQ * 256
    float* wtU   = accb  + (size_t)NQ * EMBED;           // 3072*256
    float* wtA   = wtU   + (size_t)3072 * 256;           // 1024*256
    float* wtO   = wtA   + (size_t)1024 * 256;           // 256*256
    float* wt3   = wtO   + (size_t)256 * 256;            // 16*256

    transpose_pad_kernel<<<dim3(96, 8), 256, 0, stream>>>(Wuvd,  wtU, 3072, 3072);
    transpose_pad_kernel<<<dim3(32, 8), 256, 0, stream>>>(Wattn, wtA, 1024, 1024);
    transpose_pad_kernel<<<dim3(8, 8),  256, 0, stream>>>(Wout,  wtO, 256,  256);
    transpose_pad_kernel<<<dim3(1, 8),  256, 0, stream>>>(Woff3, wt3, 12,   16);

    gemm_qproj_kernel<<<dim3(NQ / 16, 65), 128, 0, stream>>>(
        query, wtU, buvd, wtA, battn, wt3, boff3, delta, attn, off3);
    softmax_kernel<<<(NQ * HEADS + 3) / 4, 128, 0, stream>>>(attn);
    proj_kernel<<<(NCAM * NQ * NR1 + 255) / 256, 256, 0, stream>>>(
        gm, off3, l2i, Hop, Wop, ref, val);
    sample_kernel<<<NQ, 256, 0, stream>>>(
        f0, f1, f2, f3, dm0, dm1, dm2, dm3, ref, val, attn, delta, accb);
    gemm_out_kernel<<<dim3(NQ / 16, 4), 128, 0, stream>>>(accb, wtO, bout, (float*)d_out);
}